// HMSAF_13529146982747
// MI455X (gfx1250) — compile-verified
//
#include <hip/hip_runtime.h>
#include <hip/hip_bf16.h>
#include <math.h>

// H=8, N=2048, DIN=768, DOUT=1024, HD=128 — fp32 attention, HBM-bound on the
// 134MB attn_final output. All GEMMs use V_WMMA_F32_16X16X4_F32; tiles are
// staged into LDS with GLOBAL_LOAD_ASYNC_TO_LDS (ASYNCcnt) when the builtins
// exist, double-buffered so copies overlap WMMA. Stride-36 rows keep b128
// alignment and conflict-free A-fragment reads (gcd(36,64)=4).

#define HH   8
#define NN   2048
#define DIN_ 768
#define DOUT_ 1024
#define HD_  128
#define SCALE_ 0.08838834764831845f
#define NEGV (-1e9f)
#define EPS_ 1e-5f
#define LDSW 36   // LDS row stride in floats (144B: 16B aligned)

typedef float v2f __attribute__((ext_vector_type(2)));
typedef float v8f __attribute__((ext_vector_type(8)));

#if __has_builtin(__builtin_amdgcn_global_load_async_to_lds_b128) && \
    __has_builtin(__builtin_amdgcn_global_load_async_to_lds_b32) &&  \
    __has_builtin(__builtin_amdgcn_s_wait_asynccnt)
#define HAS_ASYNC_LDS 1
#else
#define HAS_ASYNC_LDS 0
#endif

#if HAS_ASYNC_LDS
typedef int vsi4 __attribute__((vector_size(16)));
typedef __attribute__((address_space(1))) vsi4 g_vsi4;  // global, 16B payload
typedef __attribute__((address_space(3))) vsi4 l_vsi4;  // LDS, 16B payload
typedef __attribute__((address_space(1))) int g_i32;
typedef __attribute__((address_space(3))) int l_i32;
#endif

// Global (AS1) keeps the 64-bit value; LDS (AS3) is the low 32 bits of the
// generic address (flat LDS aperture truncation rule).
__device__ __forceinline__ void cp_async_f4(void* l, const void* g) {
#if HAS_ASYNC_LDS
  __builtin_amdgcn_global_load_async_to_lds_b128(
      (g_vsi4*)(uintptr_t)g, (l_vsi4*)(uint32_t)(uintptr_t)l, 0, 0);
#else
  ((float*)l)[0] = ((const float*)g)[0];
  ((float*)l)[1] = ((const float*)g)[1];
  ((float*)l)[2] = ((const float*)g)[2];
  ((float*)l)[3] = ((const float*)g)[3];
#endif
}
__device__ __forceinline__ void cp_async_f1(void* l, const void* g) {
#if HAS_ASYNC_LDS
  __builtin_amdgcn_global_load_async_to_lds_b32(
      (g_i32*)(uintptr_t)g, (l_i32*)(uint32_t)(uintptr_t)l, 0, 0);
#else
  *(float*)l = *(const float*)g;
#endif
}
__device__ __forceinline__ void async_wait0() {
#if HAS_ASYNC_LDS
  __builtin_amdgcn_s_wait_asynccnt(0);
#endif
}

__device__ __forceinline__ v8f wmma4(v2f a, v2f b, v8f c) {
  return __builtin_amdgcn_wmma_f32_16x16x4_f32(false, a, false, b, (short)0, c,
                                               false, false);
}
// A fragment: 16x4; lanes 0-15 hold K={k0,k0+1}, lanes 16-31 K={k0+2,k0+3}
__device__ __forceinline__ v2f fragA(const float* As, int m0, int k0, int lane) {
  int m = m0 + (lane & 15);
  int kh = k0 + ((lane >> 4) << 1);
  v2f a;
  a.x = As[m * LDSW + kh];
  a.y = As[m * LDSW + kh + 1];
  return a;
}
// B fragment: 4x16; row K striped across lanes, K wraps across half-waves
__device__ __forceinline__ v2f fragB(const float* Bs, int k0, int n0, int lane) {
  int n = n0 + (lane & 15);
  int kh = k0 + ((lane >> 4) << 1);
  v2f b;
  b.x = Bs[kh * LDSW + n];
  b.y = Bs[(kh + 1) * LDSW + n];
  return b;
}

// ---------------- K1: x[N,DIN] @ W[DIN,DOUT] + b -> dst[H][N][HD] ----------------
__global__ __launch_bounds__(128) void qkv_gemm(const float* __restrict__ x,
                                                const float* __restrict__ w,
                                                const float* __restrict__ bias,
                                                float* __restrict__ dst) {
  __shared__ __align__(16) float As[32 * LDSW];
  __shared__ __align__(16) float Bs[32 * LDSW];
  const int col0 = blockIdx.x * 32;
  const int row0 = blockIdx.y * 32;
  const int t = threadIdx.x, lane = t & 31, wave = t >> 5;
  const int wm = (wave >> 1) * 16, wn = (wave & 1) * 16;
  v8f acc = {};
  for (int k0 = 0; k0 < DIN_; k0 += 32) {
    for (int i = t; i < 256; i += 128) {  // 256 float4 chunks per tile
      int r = i >> 3, c4 = (i & 7) * 4;
      cp_async_f4(&As[r * LDSW + c4], &x[(row0 + r) * DIN_ + k0 + c4]);
      cp_async_f4(&Bs[r * LDSW + c4], &w[(k0 + r) * DOUT_ + col0 + c4]);
    }
    async_wait0();
    __syncthreads();
#pragma unroll
    for (int kk = 0; kk < 32; kk += 4)
      acc = wmma4(fragA(As, wm, kk, lane), fragB(Bs, kk, wn, lane), acc);
    __syncthreads();
  }
  const int col = col0 + wn + (lane & 15);
  const int rbase = row0 + wm + ((lane >> 4) << 3);
  const int h = col >> 7, d = col & 127;
  const float bv = bias[col];
#pragma unroll
  for (int r = 0; r < 8; ++r) {
    int row = rbase + r;
    dst[((size_t)h * NN + row) * HD_ + d] = acc[r] + bv;
  }
}

// ---------------- K2: gates = tanh(q_flat @ gw + gb), out[N][H] ----------------
__global__ __launch_bounds__(256) void gate_kernel(const float* __restrict__ qkv,
                                                   const float* __restrict__ gw,
                                                   const float* __restrict__ gb,
                                                   float* __restrict__ out) {
  int idx = blockIdx.x * blockDim.x + threadIdx.x;  // n*H + h
  if (idx >= NN * HH) return;
  int n = idx >> 3, hsel = idx & 7;
  float acc = gb[hsel];
  for (int hp = 0; hp < HH; ++hp) {
    const float* qrow = qkv + ((size_t)hp * NN + n) * HD_;
    const float* wrow = gw + (size_t)(hp * HD_) * HH + hsel;
#pragma unroll 4
    for (int dd = 0; dd < HD_; ++dd) acc += qrow[dd] * wrow[dd * HH];
  }
  out[idx] = tanhf(acc);
}

// ---------------- K3: masked scaled scores S[h][i][j], double-buffered ----------------
__global__ __launch_bounds__(128) void scores_kernel(const float* __restrict__ q,
                                                     const float* __restrict__ k,
                                                     const int* __restrict__ batch,
                                                     float* __restrict__ S) {
  __shared__ __align__(16) float Qs[2][32 * LDSW];
  __shared__ __align__(16) float Ks[2][32 * LDSW];  // Ks[kk][j]
  const int j0 = blockIdx.x * 32;
  const int i0 = blockIdx.y * 32;
  const int h = blockIdx.z;
  const int t = threadIdx.x, lane = t & 31, wave = t >> 5;
  const int wm = (wave >> 1) * 16, wn = (wave & 1) * 16;
  const float* qh = q + (size_t)h * NN * HD_;
  const float* kh = k + (size_t)h * NN * HD_;

  auto loadQ = [&](int buf, int k0) {
    for (int i = t; i < 256; i += 128) {
      int r = i >> 3, c4 = (i & 7) * 4;
      cp_async_f4(&Qs[buf][r * LDSW + c4], &qh[(i0 + r) * HD_ + k0 + c4]);
    }
  };
  auto loadK = [&](int buf, int k0) {  // transposed scatter into LDS
    for (int i = t; i < 1024; i += 128) {
      int r = i >> 5, c = i & 31;
      cp_async_f1(&Ks[buf][c * LDSW + r], &kh[(j0 + r) * HD_ + k0 + c]);
    }
  };

  v8f acc = {};
  loadQ(0, 0);
  loadK(0, 0);
  async_wait0();
  __syncthreads();
  int buf = 0;
  for (int k0 = 0; k0 < HD_; k0 += 32) {
    if (k0 + 32 < HD_) {  // prefetch next tile into the other buffer
      loadQ(buf ^ 1, k0 + 32);
      loadK(buf ^ 1, k0 + 32);
    }
#pragma unroll
    for (int kk = 0; kk < 32; kk += 4)
      acc = wmma4(fragA(Qs[buf], wm, kk, lane), fragB(Ks[buf], kk, wn, lane), acc);
    async_wait0();
    __syncthreads();
    buf ^= 1;
  }
  const int col = j0 + wn + (lane & 15);
  const int rbase = i0 + wm + ((lane >> 4) << 3);
  const int bcol = batch[col];
#pragma unroll
  for (int r = 0; r < 8; ++r) {
    int row = rbase + r;
    float s = acc[r] * SCALE_;
    s = (batch[row] == bcol) ? s : NEGV;
    S[((size_t)h * NN + row) * NN + col] = s;
  }
}

// ---------------- K4: attn_final = S + swapT(a*S+b*motif) + gate + head-mix ----------------
__global__ __launch_bounds__(256) void attn_final_kernel(
    const float* __restrict__ S, const float* __restrict__ motif,
    const float* __restrict__ gq, const float* __restrict__ gk,
    const float* __restrict__ Wh, const float* __restrict__ alphap,
    const float* __restrict__ betap, float* __restrict__ out) {
  __shared__ float trS[16 * 17];
  __shared__ float trM[16 * 17];
  __shared__ float Ws[64];
  const int j0 = blockIdx.x * 16;
  const int i0 = blockIdx.y * 16;
  const int t = threadIdx.x;
  const int ti = t >> 4, tj = t & 15;
  if (t < 64) Ws[t] = Wh[t];
  const float alpha = alphap[0], beta = betap[0];
  float sij[HH], gqv[HH], gkv[HH];
#pragma unroll
  for (int h2 = 0; h2 < HH; ++h2)
    sij[h2] = S[((size_t)h2 * NN + i0 + ti) * NN + j0 + tj];
#pragma unroll
  for (int m = 0; m < HH; ++m) {
    gqv[m] = gq[(i0 + ti) * HH + m];
    gkv[m] = gk[(j0 + tj) * HH + m];
  }
  for (int m = 0; m < HH; ++m) {
    __syncthreads();
    trS[ti * 17 + tj] = S[((size_t)m * NN + j0 + ti) * NN + i0 + tj];
    trM[ti * 17 + tj] = motif[((size_t)m * NN + j0 + ti) * NN + i0 + tj];
    __syncthreads();
    float sji = trS[tj * 17 + ti];
    float mji = trM[tj * 17 + ti];
    float inter = 0.f;
#pragma unroll
    for (int h2 = 0; h2 < HH; ++h2) inter += sij[h2] * Ws[h2 * HH + m];
    float val = sij[m] + alpha * sji + beta * mji +
                sij[m] * (gqv[m] + gkv[m]) + inter;
    out[((size_t)m * NN + i0 + ti) * NN + j0 + tj] = val;
  }
}

// ---------------- K5: per-row online softmax stats (wave32) ----------------
__global__ __launch_bounds__(128) void softmax_stats_kernel(
    const float* __restrict__ A, float* __restrict__ rowmax,
    float* __restrict__ rowsum) {
  const int wave = threadIdx.x >> 5;
  const int lane = threadIdx.x & 31;
  const int row = blockIdx.x * 4 + wave;  // flattened h*N+i
  const float* arow = A + (size_t)row * NN;
  float m = -3.4e38f, s = 0.f;
  for (int j = lane; j < NN; j += 32) {
    __builtin_prefetch(arow + j + 256, 0, 1);
    float a = arow[j];
    float nm = fmaxf(m, a);
    s = s * __expf(m - nm) + __expf(a - nm);
    m = nm;
  }
#pragma unroll
  for (int off = 16; off > 0; off >>= 1) {
    float om = __shfl_xor(m, off, 32);
    float os = __shfl_xor(s, off, 32);
    float nm = fmaxf(m, om);
    s = s * __expf(m - nm) + os * __expf(om - nm);
    m = nm;
  }
  if (lane == 0) {
    rowmax[row] = m;
    rowsum[row] = s;
  }
}

// ---------------- K6: out = softmax(A) @ v, double-buffered, probs never hit HBM ----------------
__global__ __launch_bounds__(128) void pv_kernel(const float* __restrict__ A,
                                                 const float* __restrict__ v,
                                                 const float* __restrict__ rowmax,
                                                 const float* __restrict__ rowsum,
                                                 float* __restrict__ out) {
  __shared__ __align__(16) float Ps[2][32 * LDSW];
  __shared__ __align__(16) float Vs[2][32 * LDSW];
  const int d0 = blockIdx.x * 32;
  const int i0 = blockIdx.y * 32;
  const int h = blockIdx.z;
  const int t = threadIdx.x, lane = t & 31, wave = t >> 5;
  const int wm = (wave >> 1) * 16, wn = (wave & 1) * 16;
  const float* Ah = A + (size_t)h * NN * NN;
  const float* vh = v + (size_t)h * NN * HD_;
  const float* rmh = rowmax + h * NN;

  auto loadTiles = [&](int buf, int j0) {
    for (int i = t; i < 256; i += 128) {  // V tile: async b128
      int r = i >> 3, c4 = (i & 7) * 4;
      cp_async_f4(&Vs[buf][r * LDSW + c4], &vh[(j0 + r) * HD_ + d0 + c4]);
    }
    for (int i = t; i < 1024; i += 128) {  // P tile: exp() must go through VALU
      int r = i >> 5, c = i & 31;
      Ps[buf][r * LDSW + c] =
          __expf(Ah[(size_t)(i0 + r) * NN + j0 + c] - rmh[i0 + r]);
    }
  };

  v8f acc = {};
  loadTiles(0, 0);
  async_wait0();
  __syncthreads();
  int buf = 0;
  for (int j0 = 0; j0 < NN; j0 += 32) {
    if (j0 + 32 < NN) loadTiles(buf ^ 1, j0 + 32);
#pragma unroll
    for (int kk = 0; kk < 32; kk += 4)
      acc = wmma4(fragA(Ps[buf], wm, kk, lane), fragB(Vs[buf], kk, wn, lane), acc);
    async_wait0();
    __syncthreads();
    buf ^= 1;
  }
  const int d = d0 + wn + (lane & 15);
  const int rbase = i0 + wm + ((lane >> 4) << 3);
#pragma unroll
  for (int r = 0; r < 8; ++r) {
    int row = rbase + r;
    out[(size_t)row * DOUT_ + h * HD_ + d] = acc[r] / rowsum[h * NN + row];
  }
}

// ---------------- K7a: ao[N,DOUT] @ wo[DOUT,DIN] + bias + residual ----------------
__global__ __launch_bounds__(128) void outproj_kernel(const float* __restrict__ ao,
                                                      const float* __restrict__ wo,
                                                      const float* __restrict__ wob,
                                                      const float* __restrict__ x,
                                                      float* __restrict__ y) {
  __shared__ __align__(16) float As[32 * LDSW];
  __shared__ __align__(16) float Bs[32 * LDSW];
  const int col0 = blockIdx.x * 32;
  const int row0 = blockIdx.y * 32;
  const int t = threadIdx.x, lane = t & 31, wave = t >> 5;
  const int wm = (wave >> 1) * 16, wn = (wave & 1) * 16;
  v8f acc = {};
  for (int k0 = 0; k0 < DOUT_; k0 += 32) {
    for (int i = t; i < 256; i += 128) {
      int r = i >> 3, c4 = (i & 7) * 4;
      cp_async_f4(&As[r * LDSW + c4], &ao[(size_t)(row0 + r) * DOUT_ + k0 + c4]);
      cp_async_f4(&Bs[r * LDSW + c4], &wo[(size_t)(k0 + r) * DIN_ + col0 + c4]);
    }
    async_wait0();
    __syncthreads();
#pragma unroll
    for (int kk = 0; kk < 32; kk += 4)
      acc = wmma4(fragA(As, wm, kk, lane), fragB(Bs, kk, wn, lane), acc);
    __syncthreads();
  }
  const int col = col0 + wn + (lane & 15);
  const int rbase = row0 + wm + ((lane >> 4) << 3);
  const float bv = wob[col];
#pragma unroll
  for (int r = 0; r < 8; ++r) {
    int row = rbase + r;
    y[row * DIN_ + col] = acc[r] + bv + x[row * DIN_ + col];
  }
}

// ---------------- K7b: in-place LayerNorm over DIN=768 per row ----------------
__global__ __launch_bounds__(256) void layernorm_kernel(float* __restrict__ y,
                                                        const float* __restrict__ g,
                                                        const float* __restrict__ b) {
  __shared__ float red[256];
  const int row = blockIdx.x;
  const int t = threadIdx.x;
  float* yr = y + (size_t)row * DIN_;
  float v0 = yr[t], v1 = yr[t + 256], v2 = yr[t + 512];
  red[t] = v0 + v1 + v2;
  __syncthreads();
  for (int off = 128; off > 0; off >>= 1) {
    if (t < off) red[t] += red[t + off];
    __syncthreads();
  }
  const float mu = red[0] / (float)DIN_;
  __syncthreads();
  float d0 = v0 - mu, d1 = v1 - mu, d2 = v2 - mu;
  red[t] = d0 * d0 + d1 * d1 + d2 * d2;
  __syncthreads();
  for (int off = 128; off > 0; off >>= 1) {
    if (t < off) red[t] += red[t + off];
    __syncthreads();
  }
  const float inv = rsqrtf(red[0] / (float)DIN_ + EPS_);
  yr[t] = d0 * inv * g[t] + b[t];
  yr[t + 256] = d1 * inv * g[t + 256] + b[t + 256];
  yr[t + 512] = d2 * inv * g[t + 512] + b[t + 512];
}

extern "C" void kernel_launch(void* const* d_in, const int* in_sizes, int n_in,
                              void* d_out, int out_size, void* d_ws, size_t ws_size,
                              hipStream_t stream) {
  const float* x = (const float*)d_in[0];
  const int* batch = (const int*)d_in[1];
  const float* motif = (const float*)d_in[2];
  const float* wq_w = (const float*)d_in[3];
  const float* wq_b = (const float*)d_in[4];
  const float* wk_w = (const float*)d_in[5];
  const float* wk_b = (const float*)d_in[6];
  const float* wv_w = (const float*)d_in[7];
  const float* wv_b = (const float*)d_in[8];
  const float* wo_w = (const float*)d_in[9];
  const float* wo_b = (const float*)d_in[10];
  const float* ln_g = (const float*)d_in[11];
  const float* ln_b = (const float*)d_in[12];
  const float* alpha = (const float*)d_in[13];
  const float* beta = (const float*)d_in[14];
  const float* gq_w = (const float*)d_in[15];
  const float* gq_b = (const float*)d_in[16];
  const float* gk_w = (const float*)d_in[17];
  const float* gk_b = (const float*)d_in[18];
  const float* Wh = (const float*)d_in[19];

  float* ws = (float*)d_ws;
  float* q = ws;
  float* k = q + (size_t)HH * NN * HD_;
  float* v = k + (size_t)HH * NN * HD_;
  float* gq = v + (size_t)HH * NN * HD_;
  float* gk = gq + NN * HH;
  float* rowmax = gk + NN * HH;
  float* rowsum = rowmax + HH * NN;
  float* ao = rowsum + HH * NN;
  float* S = ao + (size_t)NN * DOUT_;

  float* y = (float*)d_out;
  float* attn = y + (size_t)NN * DIN_;

  dim3 b128(128);
  qkv_gemm<<<dim3(DOUT_ / 32, NN / 32), b128, 0, stream>>>(x, wq_w, wq_b, q);
  qkv_gemm<<<dim3(DOUT_ / 32, NN / 32), b128, 0, stream>>>(x, wk_w, wk_b, k);
  qkv_gemm<<<dim3(DOUT_ / 32, NN / 32), b128, 0, stream>>>(x, wv_w, wv_b, v);
  gate_kernel<<<dim3(NN * HH / 256), dim3(256), 0, stream>>>(q, gq_w, gq_b, gq);
  gate_kernel<<<dim3(NN * HH / 256), dim3(256), 0, stream>>>(k, gk_w, gk_b, gk);
  scores_kernel<<<dim3(NN / 32, NN / 32, HH), b128, 0, stream>>>(q, k, batch, S);
  attn_final_kernel<<<dim3(NN / 16, NN / 16), dim3(256), 0, stream>>>(
      S, motif, gq, gk, Wh, alpha, beta, attn);
  softmax_stats_kernel<<<dim3(HH * NN / 4), b128, 0, stream>>>(attn, rowmax, rowsum);
  pv_kernel<<<dim3(HD_ / 32, NN / 32, HH), b128, 0, stream>>>(attn, v, rowmax, rowsum, ao);
  outproj_kernel<<<dim3(DIN_ / 32, NN / 32), b128, 0, stream>>>(ao, wo_w, wo_b, x, y);
  layernorm_kernel<<<dim3(NN), dim3(256), 0, stream>>>(y, ln_g, ln_b);
}